// AlphasFirstIRLSStepModule_2783138808068
// MI455X (gfx1250) — compile-verified
//
#include <hip/hip_runtime.h>
#include <hip/hip_bf16.h>
#include <math.h>

typedef __attribute__((ext_vector_type(2))) float v2f;
typedef __attribute__((ext_vector_type(8))) float v8f;

#define NCORR 2000
#define DDIM  9
#define DPAD  16
#define TPB   128
#define WAVES 4
#define IRLS_EPS 1e-8f

__global__ __launch_bounds__(TPB, 1)
void irls_min_eig_fused(const float* __restrict__ G,
                        const float* __restrict__ alpha,
                        float* __restrict__ out)
{
    // LDS budget: 128000 + 8000 + 8000 + 4096 + 64 = 148160 B (< 320KB WGP limit)
    __shared__ float Ghat[NCORR * DPAD];   // sqrt(W)-scaled G, zero-padded cols 9..15
    __shared__ float s_arr[NCORR];         // sqrt(W_n)
    __shared__ float invs[NCORR];          // 1/sqrt(W_n)
    __shared__ float red[WAVES * 256];     // per-wave 16x16 partial Gram
    __shared__ float u_lds[16];            // min eigenvector

    const int b    = blockIdx.x;
    const int tid  = threadIdx.x;
    const int lane = tid & 31;
    const int wave = tid >> 5;

    const float* Gb = G     + (size_t)b * NCORR * DDIM;
    const float* ab = alpha + (size_t)b * NCORR;

    // ---- Phase 0: IRLS weights + zero padding of cols 9..15 ----
    for (int n = tid; n < NCORR; n += TPB) {
        float a = fabsf(ab[n]) + IRLS_EPS;
        s_arr[n] = rsqrtf(a);     // sqrt(W) = 1/sqrt(|a|+eps)
        invs[n]  = sqrtf(a);      // 1/sqrt(W)
        #pragma unroll
        for (int c = DDIM; c < DPAD; ++c) Ghat[n * DPAD + c] = 0.0f;
    }
    __syncthreads();

    // ---- Phase 1: stream G once from HBM, scale by sqrt(W), stage in LDS ----
    for (int idx = tid; idx < NCORR * DDIM; idx += TPB) {
        int n = idx / DDIM;
        int c = idx - n * DDIM;
        Ghat[n * DPAD + c] = Gb[idx] * s_arr[n];
    }
    __syncthreads();

    // ---- Phase 2: Gram matrix B_k = Ghat^T Ghat via V_WMMA_F32_16X16X4_F32 ----
    // A (16x4, MxK): lanes 0-15 -> VGPR0=K0,VGPR1=K1 ; lanes 16-31 -> K2,K3.
    // For a Gram update A[i][k] = Ghat[n0+k][i] and B[k][j] = Ghat[n0+k][j],
    // so both operands carry the same per-lane data.
    v8f acc = {0.f, 0.f, 0.f, 0.f, 0.f, 0.f, 0.f, 0.f};
    const int m    = lane & 15;
    const int half = lane >> 4;
    const int rows_per_wave = NCORR / WAVES;          // 500
    const int base_row = wave * rows_per_wave;
    for (int ch = 0; ch < rows_per_wave / 4; ++ch) {  // 125 WMMAs per wave
        int n0 = base_row + ch * 4 + 2 * half;
        v2f t;
        t.x = Ghat[(n0 + 0) * DPAD + m];
        t.y = Ghat[(n0 + 1) * DPAD + m];
        acc = __builtin_amdgcn_wmma_f32_16x16x4_f32(
                  /*neg_a=*/false, t, /*neg_b=*/false, t,
                  /*c_mod=*/(short)0, acc, /*reuse_a=*/false, /*reuse_b=*/false);
    }

    // spill each wave's partial 16x16 (VGPR j, lane) layout: M = j + 8*(lane>=16), N = lane&15
    #pragma unroll
    for (int j = 0; j < 8; ++j)
        red[wave * 256 + j * 32 + lane] = acc[j];
    __syncthreads();

    // ---- reduce across the 4 waves ----
    for (int e = tid; e < 256; e += TPB) {
        red[e] = red[e] + red[256 + e] + red[512 + e] + red[768 + e];
    }
    __syncthreads();

    // ---- Phase 3: cyclic Jacobi eigensolve of the 9x9 on thread 0 ----
    if (tid == 0) {
        float A[DDIM][DDIM], V[DDIM][DDIM];
        for (int i = 0; i < DDIM; ++i)
            for (int j = 0; j < DDIM; ++j) {
                int vr = (i < 8) ? i : (i - 8);
                int ln = (i < 8) ? j : (j + 16);
                A[i][j] = red[vr * 32 + ln];
                V[i][j] = (i == j) ? 1.0f : 0.0f;
            }
        for (int sweep = 0; sweep < 8; ++sweep) {
            for (int p = 0; p < DDIM - 1; ++p) {
                for (int q = p + 1; q < DDIM; ++q) {
                    float apq = A[p][q];
                    if (fabsf(apq) < 1e-20f) continue;
                    float theta = 0.5f * (A[q][q] - A[p][p]) / apq;
                    float tt = copysignf(1.0f, theta) /
                               (fabsf(theta) + sqrtf(theta * theta + 1.0f));
                    float cth = rsqrtf(1.0f + tt * tt);
                    float sth = tt * cth;
                    for (int k = 0; k < DDIM; ++k) {
                        float akp = A[k][p], akq = A[k][q];
                        A[k][p] = cth * akp - sth * akq;
                        A[k][q] = sth * akp + cth * akq;
                    }
                    for (int k = 0; k < DDIM; ++k) {
                        float apk = A[p][k], aqk = A[q][k];
                        A[p][k] = cth * apk - sth * aqk;
                        A[q][k] = sth * apk + cth * aqk;
                    }
                    for (int k = 0; k < DDIM; ++k) {
                        float vkp = V[k][p], vkq = V[k][q];
                        V[k][p] = cth * vkp - sth * vkq;
                        V[k][q] = sth * vkp + cth * vkq;
                    }
                }
            }
        }
        int   mi = 0;
        float mv = A[0][0];
        for (int i = 1; i < DDIM; ++i)
            if (A[i][i] < mv) { mv = A[i][i]; mi = i; }
        for (int i = 0; i < DDIM; ++i) u_lds[i] = V[i][mi];
    }
    __syncthreads();

    // ---- Phase 4: out[n] = | (Ghat[n,:] . u) / sqrt(W_n) |  (== |G[n,:] . u|) ----
    float u[DDIM];
    #pragma unroll
    for (int c = 0; c < DDIM; ++c) u[c] = u_lds[c];
    for (int n = tid; n < NCORR; n += TPB) {
        const float* row = &Ghat[n * DPAD];
        float d = 0.0f;
        #pragma unroll
        for (int c = 0; c < DDIM; ++c) d = fmaf(row[c], u[c], d);
        out[(size_t)b * NCORR + n] = fabsf(d * invs[n]);
    }
}

extern "C" void kernel_launch(void* const* d_in, const int* in_sizes, int n_in,
                              void* d_out, int out_size, void* d_ws, size_t ws_size,
                              hipStream_t stream) {
    const float* G     = (const float*)d_in[0];
    const float* alpha = (const float*)d_in[1];
    float* out = (float*)d_out;
    int B = in_sizes[1] / NCORR;   // alpha_k is [B, N]
    irls_min_eig_fused<<<B, TPB, 0, stream>>>(G, alpha, out);
}